// PlanarNet_33732673143510
// MI455X (gfx1250) — compile-verified
//
#include <hip/hip_runtime.h>

#define B_  64
#define D_  256
#define E_  512
#define ND_ 1024
#define NB_ 32

typedef __attribute__((ext_vector_type(2))) float v2f;
typedef __attribute__((ext_vector_type(8))) float v8f;

// ---------------------------------------------------------------------------
// Kernel 1: w_dir[b, 2e] = w_dir[b, 2e+1] = sign(b,e) * p_e/(1-p_e)
// sign = +1 if parity(det[b,:] & pebz[:,e]) == 0 else -1
// ---------------------------------------------------------------------------
__global__ __launch_bounds__(256) void kw_wdir_kernel(const int* __restrict__ det,
                                                      const int* __restrict__ pebz,
                                                      const float* __restrict__ para,
                                                      float* __restrict__ wdir)
{
    const int b = blockIdx.x;
    const int* dr = det + (size_t)b * D_;
    for (int e = threadIdx.x; e < E_; e += 256) {
        int acc = 0;
        for (int d = 0; d < D_; ++d)
            acc ^= dr[d] & pebz[(size_t)d * E_ + e];
        float x = para[e];
        float p = 1.0f / (1.0f + expf(-x)) + 1e-20f;
        float w = p / (1.0f - p);
        float v = (acc & 1) ? -w : w;
        wdir[(size_t)b * ND_ + 2 * e]     = v;
        wdir[(size_t)b * ND_ + 2 * e + 1] = v;
    }
}

// ---------------------------------------------------------------------------
// Kernel 2: M_b[row,col] = (row==col) - kwz[row,col] * w_dir[b,col]
// grid.x = nb * (ND*ND/256)
// ---------------------------------------------------------------------------
__global__ __launch_bounds__(256) void kw_buildM_kernel(const float* __restrict__ kwz,
                                                        const float* __restrict__ wdir,
                                                        float* __restrict__ Mbuf,
                                                        int batch0)
{
    const size_t per = (size_t)ND_ * ND_;
    size_t idx = (size_t)blockIdx.x * 256 + threadIdx.x;
    int    lb  = (int)(idx / per);
    size_t r   = idx - (size_t)lb * per;
    int    row = (int)(r >> 10);          // r / ND_
    int    col = (int)(r & (ND_ - 1));    // r % ND_
    float v = -kwz[r] * wdir[(size_t)(batch0 + lb) * ND_ + col];
    if (row == col) v += 1.0f;
    Mbuf[idx] = v;
}

// ---------------------------------------------------------------------------
// Kernel 3: in-place unpivoted blocked LU of one 1024x1024 matrix per block,
// accumulating log|det| = sum log|U_jj|.  Trailing update uses f32 WMMA with
// 2x2 register blocking (32x32 C per wave).  256 threads = 8 wave32 / block.
// ---------------------------------------------------------------------------
__global__ __launch_bounds__(256) void kw_lu_kernel(float* __restrict__ Mbase,
                                                    float* __restrict__ ldet,
                                                    int batch0)
{
    float* __restrict__ M = Mbase + (size_t)blockIdx.x * ND_ * ND_;
    const int tid  = threadIdx.x;
    const int lane = tid & 31;
    const int wave = tid >> 5;            // 0..7

    __shared__ float urow[NB_];           // pivot-row segment of current panel
    __shared__ float L11[NB_][NB_ + 1];   // diagonal block (unit lower) for trsm
    __shared__ float s_ldet;
    if (tid == 0) s_ldet = 0.0f;
    __syncthreads();

    for (int k0 = 0; k0 < ND_; k0 += NB_) {
        // ---- panel factorization: cols [k0, k0+NB), rows [k0, ND), in place --
        for (int j = 0; j < NB_; ++j) {
            const int pr = k0 + j;
            if (tid < NB_) urow[tid] = M[(size_t)pr * ND_ + k0 + tid];
            __syncthreads();
            const float piv = urow[j];
            if (tid == 0) s_ldet += logf(fabsf(piv));
            const float inv = 1.0f / piv;
            for (int r = pr + 1 + tid; r < ND_; r += 256) {
                float* row = M + (size_t)r * ND_ + k0;
                float l = row[j] * inv;
                row[j] = l;
                for (int c = j + 1; c < NB_; ++c) row[c] -= l * urow[c];
            }
            __syncthreads();
        }
        const int T0 = k0 + NB_;          // trailing offset
        if (T0 >= ND_) break;

        // ---- stage L11 (unit lower diag block) into LDS ----------------------
        for (int t = tid; t < NB_ * NB_; t += 256) {
            int r = t >> 5, c = t & 31;
            L11[r][c] = M[(size_t)(k0 + r) * ND_ + k0 + c];
        }
        __syncthreads();

        // ---- U12 = L11^{-1} * A12 (forward substitution per column) ---------
        for (int c = T0 + tid; c < ND_; c += 256) {
            float u[NB_];
            #pragma unroll
            for (int i = 0; i < NB_; ++i) {
                float a = M[(size_t)(k0 + i) * ND_ + c];
                #pragma unroll
                for (int t2 = 0; t2 < i; ++t2) a -= L11[i][t2] * u[t2];
                u[i] = a;
            }
            #pragma unroll
            for (int i = 0; i < NB_; ++i) M[(size_t)(k0 + i) * ND_ + c] = u[i];
        }
        __syncthreads();

        // ---- trailing update: A22 -= L21 * U12 via V_WMMA_F32_16X16X4_F32 ----
        // Each wave owns a 32x32 C block = 2x2 WMMA tiles; K=32 as 8 K=4 steps;
        // 32 WMMAs per block with 2x reuse of every A and B fragment.
        // f32 WMMA has no A/B negation (NEG = CNeg only), so load A as -L21.
        {
            const int TS   = ND_ - T0;            // multiple of 32
            const int tn2  = TS >> 5;             // 32-wide tile grid
            const int nt   = tn2 * tn2;
            const int half = lane >> 4;           // 0 | 1
            const int lidx = lane & 15;
            for (int t = wave; t < nt; t += 8) {
                const int tm = t / tn2, tn = t - tm * tn2;
                const int m0 = T0 + tm * 32, n0 = T0 + tn * 32;
                // C tiles: c[mi][ni] covers rows m0+16*mi.., cols n0+16*ni..
                v8f c00, c01, c10, c11;
                #pragma unroll
                for (int r = 0; r < 8; ++r) {     // C: VGPR r -> M=r+8*half, N=lidx
                    const size_t ro0 = (size_t)(m0 + half * 8 + r) * ND_;
                    const size_t ro1 = (size_t)(m0 + 16 + half * 8 + r) * ND_;
                    c00[r] = M[ro0 + n0 + lidx];
                    c01[r] = M[ro0 + n0 + 16 + lidx];
                    c10[r] = M[ro1 + n0 + lidx];
                    c11[r] = M[ro1 + n0 + 16 + lidx];
                }
                #pragma unroll
                for (int kk = 0; kk < NB_; kk += 4) {
                    const int ka = k0 + kk + half * 2;
                    v2f a0, a1, b0, b1;
                    // A 16x4: lanes 0-15 K={0,1}, lanes 16-31 K={2,3}; M=lidx
                    const size_t ar0 = (size_t)(m0 + lidx) * ND_ + ka;
                    const size_t ar1 = (size_t)(m0 + 16 + lidx) * ND_ + ka;
                    a0.x = -M[ar0];     a0.y = -M[ar0 + 1];
                    a1.x = -M[ar1];     a1.y = -M[ar1 + 1];
                    // B 4x16: vgpr0 K={0|2}, vgpr1 K={1|3}; N=lidx
                    const size_t br0 = (size_t)ka * ND_;
                    const size_t br1 = (size_t)(ka + 1) * ND_;
                    b0.x = M[br0 + n0 + lidx];       b0.y = M[br1 + n0 + lidx];
                    b1.x = M[br0 + n0 + 16 + lidx];  b1.y = M[br1 + n0 + 16 + lidx];
                    c00 = __builtin_amdgcn_wmma_f32_16x16x4_f32(
                              false, a0, false, b0, (short)0, c00, false, false);
                    c01 = __builtin_amdgcn_wmma_f32_16x16x4_f32(
                              false, a0, false, b1, (short)0, c01, false, false);
                    c10 = __builtin_amdgcn_wmma_f32_16x16x4_f32(
                              false, a1, false, b0, (short)0, c10, false, false);
                    c11 = __builtin_amdgcn_wmma_f32_16x16x4_f32(
                              false, a1, false, b1, (short)0, c11, false, false);
                }
                #pragma unroll
                for (int r = 0; r < 8; ++r) {
                    const size_t ro0 = (size_t)(m0 + half * 8 + r) * ND_;
                    const size_t ro1 = (size_t)(m0 + 16 + half * 8 + r) * ND_;
                    M[ro0 + n0 + lidx]      = c00[r];
                    M[ro0 + n0 + 16 + lidx] = c01[r];
                    M[ro1 + n0 + lidx]      = c10[r];
                    M[ro1 + n0 + 16 + lidx] = c11[r];
                }
            }
        }
        __syncthreads();
    }

    __syncthreads();
    if (tid == 0) ldet[batch0 + blockIdx.x] = s_ldet;
}

// ---------------------------------------------------------------------------
// Kernel 4: loss = -( sum_e log1p(-p_e) + 0.5 * mean_b logdet_b )
// ---------------------------------------------------------------------------
__global__ __launch_bounds__(256) void kw_finalize_kernel(const float* __restrict__ para,
                                                          const float* __restrict__ ldet,
                                                          float* __restrict__ out)
{
    __shared__ float red[256];
    const int tid = threadIdx.x;
    float s = 0.0f;
    for (int e = tid; e < E_; e += 256) {
        float p = 1.0f / (1.0f + expf(-para[e])) + 1e-20f;
        s += log1pf(-p);
    }
    red[tid] = s;
    __syncthreads();
    for (int off = 128; off > 0; off >>= 1) {
        if (tid < off) red[tid] += red[tid + off];
        __syncthreads();
    }
    const float base = red[0];
    __syncthreads();
    float t = 0.0f;
    for (int b = tid; b < B_; b += 256) t += ldet[b];
    red[tid] = t;
    __syncthreads();
    for (int off = 128; off > 0; off >>= 1) {
        if (tid < off) red[tid] += red[tid + off];
        __syncthreads();
    }
    if (tid == 0) out[0] = -(base + 0.5f * red[0] / (float)B_);
}

// ---------------------------------------------------------------------------
extern "C" void kernel_launch(void* const* d_in, const int* in_sizes, int n_in,
                              void* d_out, int out_size, void* d_ws, size_t ws_size,
                              hipStream_t stream)
{
    (void)in_sizes; (void)n_in; (void)out_size;
    const int*   det  = (const int*)  d_in[0];   // [B, D]
    const int*   pebz = (const int*)  d_in[1];   // [D, E]
    const float* para = (const float*)d_in[2];   // [E]
    const float* kwz  = (const float*)d_in[3];   // [ND, ND]
    // d_in[4] (edges_dict_z) encodes j//2 — folded into indexing, not needed.

    // Workspace layout: [ w_dir: B*ND f32 | ldet: B f32 | pad | M chunk buffer ]
    float* wdir = (float*)d_ws;
    float* ldet = wdir + (size_t)B_ * ND_;
    size_t head = ((size_t)B_ * ND_ + B_) * sizeof(float);
    head = (head + 255) & ~(size_t)255;
    float* Mbuf = (float*)((char*)d_ws + head);

    const size_t perM = (size_t)ND_ * ND_ * sizeof(float);   // 4 MB / matrix
    size_t avail = (ws_size > head) ? (ws_size - head) : 0;
    int nchunk = (int)(avail / perM);
    if (nchunk < 1)  nchunk = 1;
    if (nchunk > B_) nchunk = B_;

    kw_wdir_kernel<<<B_, 256, 0, stream>>>(det, pebz, para, wdir);

    const unsigned blocksPerM = (unsigned)((size_t)ND_ * ND_ / 256);  // 4096
    for (int c0 = 0; c0 < B_; c0 += nchunk) {
        int nb = (B_ - c0 < nchunk) ? (B_ - c0) : nchunk;
        kw_buildM_kernel<<<blocksPerM * (unsigned)nb, 256, 0, stream>>>(kwz, wdir, Mbuf, c0);
        kw_lu_kernel<<<nb, 256, 0, stream>>>(Mbuf, ldet, c0);
    }

    kw_finalize_kernel<<<1, 256, 0, stream>>>(para, ldet, (float*)d_out);
}